// CAM_50483045597848
// MI455X (gfx1250) — compile-verified
//
#include <hip/hip_runtime.h>
#include <hip/hip_bf16.h>

typedef float v2f __attribute__((ext_vector_type(2)));
typedef float v8f __attribute__((ext_vector_type(8)));

#define Bn   8
#define Cn   64
#define Nn   65536             // 256*256
#define SLICES 64
#define KCHUNK (Nn / SLICES)   // 1024

// ---------------------------------------------------------------------------
// Kernel 1: split-K Gram partials via V_WMMA_F32_16X16X4_F32, fused row sums.
// grid = (SLICES, B), block = 512 (16 waves). Wave w -> tile (w>>2, w&3).
// partial[b][s][64][64] = X_chunk X_chunkᵀ  (raw x, no centering)
// psum[b][s][64]        = per-chunk row sums (from diagonal waves, free reuse
//                         of the A-fragment loads -> x is read only twice total)
// ---------------------------------------------------------------------------
__global__ __launch_bounds__(512) void cam_gram_kernel(const float* __restrict__ x,
                                                       float* __restrict__ partial,
                                                       float* __restrict__ psum) {
    const int s    = blockIdx.x;
    const int b    = blockIdx.y;
    const int wave = threadIdx.x >> 5;
    const int lane = threadIdx.x & 31;
    const int ti   = wave >> 2;          // A row-block 0..3
    const int tj   = wave & 3;           // B row-block 0..3
    const int half = lane >> 4;          // selects K pair (0,1) vs (2,3)
    const int l    = lane & 15;

    const size_t base = (size_t)b * Cn * Nn;
    const float* Arow = x + base + (size_t)(ti * 16 + l) * Nn;   // A: M = ti*16+l
    const float* Brow = x + base + (size_t)(tj * 16 + l) * Nn;   // B: N = tj*16+l (Xᵀ tile)
    const int k0 = s * KCHUNK;

    v8f acc = {};
    float rsum = 0.f;                    // partial row sum (used by diagonal waves)
    #pragma unroll 4
    for (int k = k0; k < k0 + KCHUNK; k += 4) {
        const int kk = k + 2 * half;
        const float2 av = *(const float2*)(Arow + kk);
        const float2 bv = *(const float2*)(Brow + kk);
        rsum += av.x + av.y;             // branch-free; negligible vs WMMA
        v2f a  = {av.x, av.y};
        v2f bb = {bv.x, bv.y};
        acc = __builtin_amdgcn_wmma_f32_16x16x4_f32(
                  false, a, false, bb, (short)0, acc, false, false);
    }

    float* out = partial + (size_t)(b * SLICES + s) * (Cn * Cn);
    #pragma unroll
    for (int r = 0; r < 8; ++r) {
        const int M = r + 8 * half;      // C/D layout: vgpr r = rows r and r+8
        out[(ti * 16 + M) * Cn + tj * 16 + l] = acc[r];
    }

    if (ti == tj) {                      // wave-uniform branch (waves 0,5,10,15)
        // lanes l and l^16 cover K pairs (0,1) and (2,3): combine for full sum
        const float other = __shfl(rsum, lane ^ 16, 32);
        if (half == 0)
            psum[(size_t)(b * SLICES + s) * Cn + ti * 16 + l] = rsum + other;
    }
}

// ---------------------------------------------------------------------------
// Kernel 2: reduce split-K partials + row sums, mean-correct, softmax, rowdot.
// grid = B, block = 256.
// ---------------------------------------------------------------------------
__global__ __launch_bounds__(256) void cam_softmax_kernel(const float* __restrict__ partial,
                                                          const float* __restrict__ psum,
                                                          float* __restrict__ att,
                                                          float* __restrict__ rowdot) {
    const int b = blockIdx.x;
    __shared__ float E[Cn * Cn];
    __shared__ float m[Cn];
    const float* pb = partial + (size_t)b * SLICES * (Cn * Cn);
    const float* ps = psum + (size_t)b * SLICES * Cn;

    // channel means from per-slice row sums (fixed order -> deterministic)
    if (threadIdx.x < Cn) {
        float sum = 0.f;
        for (int sl = 0; sl < SLICES; ++sl) sum += ps[sl * Cn + threadIdx.x];
        m[threadIdx.x] = sum * (1.0f / (float)Nn);
    }
    __syncthreads();

    for (int e = threadIdx.x; e < Cn * Cn; e += 256) {
        float sum = 0.f;
        for (int sl = 0; sl < SLICES; ++sl)
            sum += pb[(size_t)sl * (Cn * Cn) + e];
        const int c = e >> 6, d = e & 63;
        // energy = (Σ x_c x_d)/N - m_c m_d  ==  centered covariance / N
        E[e] = sum * (1.0f / (float)Nn) - m[c] * m[d];
    }
    __syncthreads();

    if (threadIdx.x < Cn) {
        const int c = threadIdx.x;
        float* row = E + c * Cn;         // this thread owns the row exclusively
        float mx = row[0];
        for (int d = 1; d < Cn; ++d) mx = fmaxf(mx, row[d]);
        float ssum = 0.f;
        for (int d = 0; d < Cn; ++d) { const float e = __expf(row[d] - mx); row[d] = e; ssum += e; }
        const float inv = 1.0f / ssum;
        float rd = 0.f;
        float* ao = att + (size_t)b * (Cn * Cn) + c * Cn;
        for (int d = 0; d < Cn; ++d) {
            const float a = row[d] * inv;
            ao[d] = a;
            rd += a * m[d];
        }
        rowdot[b * Cn + c] = rd;         // Σ_d att[c,d] * mean[d]
    }
}

// ---------------------------------------------------------------------------
// Kernel 3: out[b,c,n] = gamma * ( Σ_d att[b,c,d] x[b,d,n]  -  rowdot[b,c] )
// grid = (N/32, B), block = 256 (8 waves): wave -> (row tile 0..3, col half 0..1)
// 16x16 output tile per wave, K=64 via 16 WMMA f32 16x16x4 steps.
// ---------------------------------------------------------------------------
__global__ __launch_bounds__(256) void cam_out_kernel(const float* __restrict__ x,
                                                      const float* __restrict__ att,
                                                      const float* __restrict__ rowdot,
                                                      const float* __restrict__ gamma,
                                                      float* __restrict__ out) {
    const int b    = blockIdx.y;
    const int wave = threadIdx.x >> 5;
    const int lane = threadIdx.x & 31;
    const int ti   = wave >> 1;                       // row tile 0..3
    const int n0   = blockIdx.x * 32 + (wave & 1) * 16;
    const int half = lane >> 4;
    const int l    = lane & 15;

    const float g = gamma[0];
    const float* ab = att + (size_t)b * (Cn * Cn);
    const float* xb = x + (size_t)b * Cn * Nn;

    v8f acc = {};
    #pragma unroll
    for (int k = 0; k < Cn; k += 4) {
        const int kk = k + 2 * half;
        // A fragment: rows of attention tile ti, K pair (kk, kk+1)
        const float2 av = *(const float2*)(ab + (ti * 16 + l) * Cn + kk);
        v2f a = {av.x, av.y};
        // B fragment: B[k',n] = x[kk+k'][n0+n]  (two strided, lane-coalesced b32 loads)
        v2f bb = {xb[(size_t)kk * Nn + n0 + l],
                  xb[(size_t)(kk + 1) * Nn + n0 + l]};
        acc = __builtin_amdgcn_wmma_f32_16x16x4_f32(
                  false, a, false, bb, (short)0, acc, false, false);
    }

    float* ob = out + (size_t)b * Cn * Nn;
    const float* rdb = rowdot + b * Cn;
    #pragma unroll
    for (int r = 0; r < 8; ++r) {
        const int row = ti * 16 + r + 8 * half;       // C/D layout
        ob[(size_t)row * Nn + n0 + l] = g * (acc[r] - rdb[row]);
    }
}

// ---------------------------------------------------------------------------
extern "C" void kernel_launch(void* const* d_in, const int* in_sizes, int n_in,
                              void* d_out, int out_size, void* d_ws, size_t ws_size,
                              hipStream_t stream) {
    const float* x     = (const float*)d_in[0];
    const float* gamma = (const float*)d_in[1];
    float* out = (float*)d_out;

    // Workspace layout (floats):
    float* ws      = (float*)d_ws;
    float* rowdot  = ws;                                  // 512
    float* att     = ws + 512;                            // 8*4096 = 32768
    float* psum    = ws + 512 + Bn * Cn * Cn;             // 8*64*64 = 32768
    float* partial = ws + 512 + 2 * Bn * Cn * Cn + Bn * SLICES * Cn - Bn * Cn * Cn;
    // (= ws + 512 + 32768 + 32768): 8*64*4096 floats (~8 MB)
    partial = ws + 512 + 32768 + 32768;

    cam_gram_kernel<<<dim3(SLICES, Bn), 512, 0, stream>>>(x, partial, psum);
    cam_softmax_kernel<<<Bn, 256, 0, stream>>>(partial, psum, att, rowdot);
    cam_out_kernel<<<dim3(Nn / 32, Bn), 256, 0, stream>>>(x, att, rowdot, gamma, out);
}